// Multihead_Attention_21646635172230
// MI455X (gfx1250) — compile-verified
//
#include <hip/hip_runtime.h>
#include <hip/hip_bf16.h>
#include <cstdint>

typedef float v2f __attribute__((ext_vector_type(2)));
typedef float v4f __attribute__((ext_vector_type(4)));
typedef float v8f __attribute__((ext_vector_type(8)));

#define WMMA_F32(a, b, c) \
  __builtin_amdgcn_wmma_f32_16x16x4_f32(false, (a), false, (b), (short)0, (c), false, false)

constexpr int Bb  = 2;
constexpr int S   = 2048;
constexpr int DIN = 1024;
constexpr int E   = 1024;
constexpr int H   = 16;
constexpr int HD  = 64;          // head dim
constexpr int N3  = 3 * E;       // 3072
constexpr int KT  = 64;          // GEMM K tile
constexpr int XST = KT + 4;      // X tile LDS stride (272B: 16B-aligned, conflict-free)

// Async global->LDS copies (CDNA5, ASYNCcnt-tracked). LDS offset = low 32 bits
// of the flat LDS pointer (see ISA 10.2).
__device__ __forceinline__ void async_b128(uint32_t lds_off, const void* g) {
  asm volatile("global_load_async_to_lds_b128 %0, %1, off"
               :: "v"(lds_off), "v"(g) : "memory");
}
__device__ __forceinline__ void async_b32(uint32_t lds_off, const void* g) {
  asm volatile("global_load_async_to_lds_b32 %0, %1, off"
               :: "v"(lds_off), "v"(g) : "memory");
}
__device__ __forceinline__ uint32_t lds_off(const void* p) {
  return (uint32_t)(uintptr_t)p;
}

// ---------------------------------------------------------------------------
// Kernel 1: fused QKV projection  y = x @ W0 + b0, scattered into
//   Qb  [bh][s][dd]                 (natural, A-operand friendly)
//   Ktp [bh][dd/2][s]{dd,dd+1}      (d-major pair-interleaved -> B frag = 1 b64)
//   Vp  [bh][s/2][dd]{s,s+1}        (s-pair interleaved       -> B frag = 1 b64)
// Block tile 128M x 64N, K staged 64 wide, DOUBLE-BUFFERED async global->LDS,
// software-pipelined fragment loads (next kk preloaded before current WMMAs).
// ---------------------------------------------------------------------------
__global__ __launch_bounds__(256) void qkv_gemm(
    const float* __restrict__ x, const float* __restrict__ W0,
    const float* __restrict__ b0,
    float* __restrict__ Qb, float* __restrict__ Ktp, float* __restrict__ Vp)
{
  __shared__ float xs[2][128 * XST];  // 128 rows x 64 cols (stride 68)
  __shared__ float wsp[2][32 * 128];  // 32 k-pairs x 64 cols x 2 (pair-interleaved)

  const int tid  = threadIdx.x;
  const int wave = tid >> 5;
  const int lane = tid & 31;
  const int l16  = lane & 15;
  const int hi   = lane >> 4;      // K-pair selector within the wave
  const int m0   = blockIdx.y * 128;
  const int n0   = blockIdx.x * 64;

  v8f acc[4] = {};

  // issue one K-tile's copies: 8x b128 (X tile) + 16x b32 (W interleave scatter)
  auto stage = [&](int k0, int buf) {
#pragma unroll
    for (int i = 0; i < 8; ++i) {
      int idx = tid + i * 256;
      int row = idx >> 4, c4 = (idx & 15) << 2;
      async_b128(lds_off(&xs[buf][row * XST + c4]),
                 &x[(size_t)(m0 + row) * DIN + k0 + c4]);
    }
#pragma unroll
    for (int i = 0; i < 16; ++i) {
      int idx = tid + i * 256;
      int k = idx >> 6, col = idx & 63;
      async_b32(lds_off(&wsp[buf][(k >> 1) * 128 + col * 2 + (k & 1)]),
                &W0[(size_t)(k0 + k) * N3 + n0 + col]);
    }
  };

  stage(0, 0);
  for (int t = 0; t < DIN / KT; ++t) {
    const int buf = t & 1;
    if (t < DIN / KT - 1) {
      stage((t + 1) * KT, buf ^ 1);                     // prefetch next tile
      asm volatile("s_wait_asynccnt 0x18" ::: "memory"); // current tile done (<=24 left)
    } else {
      asm volatile("s_wait_asynccnt 0x0" ::: "memory");
    }
    __syncthreads();

    const float* xrow = &xs[buf][(wave * 16 + l16) * XST];
    const int    colo = l16 * 2;

    // software pipeline: preload kk=0 fragments
    v2f a_c = *reinterpret_cast<const v2f*>(&xrow[2 * hi]);
    v2f b_c[4];
#pragma unroll
    for (int nb = 0; nb < 4; ++nb)
      b_c[nb] = *reinterpret_cast<const v2f*>(&wsp[buf][hi * 128 + nb * 32 + colo]);

#pragma unroll
    for (int kk = 0; kk < KT; kk += 4) {
      const int kn = (kk + 4) & (KT - 1);               // wrap: last preload unused
      v2f a_n = *reinterpret_cast<const v2f*>(&xrow[kn + 2 * hi]);
      v2f b_n[4];
      const int pn = (kn >> 1) + hi;
#pragma unroll
      for (int nb = 0; nb < 4; ++nb)
        b_n[nb] = *reinterpret_cast<const v2f*>(&wsp[buf][pn * 128 + nb * 32 + colo]);
#pragma unroll
      for (int nb = 0; nb < 4; ++nb)
        acc[nb] = WMMA_F32(a_c, b_c[nb], acc[nb]);
      a_c = a_n;
#pragma unroll
      for (int nb = 0; nb < 4; ++nb) b_c[nb] = b_n[nb];
    }
    __syncthreads();   // all waves done reading buf before it is re-staged
  }

  // epilogue: bias + scatter into head-sliced Q / Ktp / Vp
  // (reshape(b,s,H,192): head h owns cols [h*192,h*192+192); 192 = 3*64 so each
  //  64-wide N tile is purely Q, K, or V of one head)
  const int hgrp = n0 / 192;
  const int typ  = (n0 / 64) % 3;      // 0=Q, 1=K, 2=V
  const int bidx = m0 / S;
  const int bh   = bidx * H + hgrp;

#pragma unroll
  for (int nb = 0; nb < 4; ++nb) {
    int   n    = n0 + nb * 16 + l16;
    float bias = b0[n];
    int   dd   = n - hgrp * 192 - typ * 64;   // 0..63 within Q/K/V slice
#pragma unroll
    for (int r = 0; r < 8; ++r) {
      int   m   = m0 + wave * 16 + r + 8 * hi; // D layout: M = r + 8*hi
      int   s   = m & (S - 1);
      float val = acc[nb][r] + bias;
      if (typ == 0)
        Qb[((size_t)bh * S + s) * HD + dd] = val;
      else if (typ == 1)
        Ktp[(((size_t)bh * 32 + (dd >> 1)) * S + s) * 2 + (dd & 1)] = val;
      else
        Vp[(((size_t)bh * (S / 2) + (s >> 1)) * HD + dd) * 2 + (s & 1)] = val;
    }
  }
}

// ---------------------------------------------------------------------------
// Kernel 2: causal flash attention per (b,h). One wave owns a 16-row q tile,
// 8 waves per block -> 128 q rows per block. Online softmax in fp32; both
// matmuls use V_WMMA_F32_16X16X4_F32. QK^T uses two independent 8-deep WMMA
// chains (sc0/sc1) to halve XDL dependency depth; K/V fragment loads are
// batched ahead of their WMMA chains. P moves D-layout -> A-layout through a
// padded per-wave LDS tile. Output stored as the reference's flat
// [B][H][S][64] bytes (reshape without transpose-back).
// ---------------------------------------------------------------------------
__global__ __launch_bounds__(256) void attn(
    const float* __restrict__ Qb, const float* __restrict__ Ktp,
    const float* __restrict__ Vp, float* __restrict__ out)
{
  __shared__ float pbuf[8][16 * 18];   // per-wave P tile, stride 18 (aligned + conflict-free)

  const int tid  = threadIdx.x;
  const int wave = tid >> 5;
  const int lane = tid & 31;
  const int l16  = lane & 15;
  const int hi   = lane >> 4;
  const int bh   = blockIdx.x >> 4;          // 0..31 (b*H + h)
  const int qb   = blockIdx.x & 15;
  const int q0   = qb * 128 + wave * 16;
  const float scale = 0.03125f;              // 1/sqrt(D_IN) = 1/32 (as in source)

  // Q fragments for all 16 K-steps over d=64 (one b64 load each)
  v2f aq[16];
  const float* qrow = &Qb[((size_t)bh * S + q0 + l16) * HD];
#pragma unroll
  for (int j = 0; j < 16; ++j)
    aq[j] = *reinterpret_cast<const v2f*>(&qrow[4 * j + 2 * hi]);

  v8f o[4] = {};
  float mrow[8], lrow[8];
#pragma unroll
  for (int r = 0; r < 8; ++r) { mrow[r] = -__builtin_inff(); lrow[r] = 0.0f; }

  float* pb = &pbuf[wave][0];
  const int ktmax = q0 >> 4;                 // causal: last tile is the diagonal block

  for (int kt = 0; kt <= ktmax; ++kt) {
    const int k0 = kt << 4;

    // ---- S_tile = Q (16x64) * K^T (64x16): batched loads, 2x8 WMMA chains
    const float* kbase = &Ktp[(((size_t)bh * 32 + hi) * S + k0 + l16) * 2];
    v2f bk[16];
#pragma unroll
    for (int j = 0; j < 16; ++j)
      bk[j] = *reinterpret_cast<const v2f*>(&kbase[(size_t)(2 * j) * S * 2]);

    v8f sc0 = {}, sc1 = {};
#pragma unroll
    for (int j = 0; j < 8; ++j) {
      sc0 = WMMA_F32(aq[j],     bk[j],     sc0);
      sc1 = WMMA_F32(aq[j + 8], bk[j + 8], sc1);
    }
    v8f sc = sc0 + sc1;

    // ---- scale + causal mask + online softmax (row stats over 16-lane half)
    const int kc = k0 + l16;
#pragma unroll
    for (int r = 0; r < 8; ++r) {
      int   q  = q0 + r + 8 * hi;
      float sv = sc[r] * scale;
      sv = (kc > q) ? -__builtin_inff() : sv;
      float tm = sv;
#pragma unroll
      for (int msk = 1; msk < 16; msk <<= 1) tm = fmaxf(tm, __shfl_xor(tm, msk, 32));
      float mn    = fmaxf(mrow[r], tm);
      float alpha = __expf(mrow[r] - mn);
      float p     = __expf(sv - mn);
      float rs    = p;
#pragma unroll
      for (int msk = 1; msk < 16; msk <<= 1) rs += __shfl_xor(rs, msk, 32);
      lrow[r] = lrow[r] * alpha + rs;
      mrow[r] = mn;
#pragma unroll
      for (int nb = 0; nb < 4; ++nb) o[nb][r] *= alpha;
      pb[(r + 8 * hi) * 18 + l16] = p;       // D-layout -> LDS
    }
    __builtin_amdgcn_wave_barrier();
    asm volatile("s_wait_dscnt 0x0" ::: "memory");  // same-wave LDS RAW fence

    // ---- O += P (16x16) * V (16x64): batched loads, 4 independent chains
    v2f ap[4], bv[4][4];
#pragma unroll
    for (int j2 = 0; j2 < 4; ++j2) {
      ap[j2] = *reinterpret_cast<const v2f*>(&pb[l16 * 18 + 4 * j2 + 2 * hi]);
      size_t vrow = ((size_t)bh * (S / 2) + (k0 >> 1) + 2 * j2 + hi) * HD;
#pragma unroll
      for (int nb = 0; nb < 4; ++nb)
        bv[j2][nb] = *reinterpret_cast<const v2f*>(&Vp[(vrow + nb * 16 + l16) * 2]);
    }
#pragma unroll
    for (int j2 = 0; j2 < 4; ++j2)
#pragma unroll
      for (int nb = 0; nb < 4; ++nb)
        o[nb] = WMMA_F32(ap[j2], bv[j2][nb], o[nb]);
    __builtin_amdgcn_wave_barrier();
  }

  // normalize and store in [B][H][S][64] flat order (matches o.reshape(b,s,-1))
#pragma unroll
  for (int r = 0; r < 8; ++r) {
    float inv = 1.0f / lrow[r];
    int   q   = q0 + r + 8 * hi;
    float* orow = &out[((size_t)bh * S + q) * HD];
#pragma unroll
    for (int nb = 0; nb < 4; ++nb)
      orow[nb * 16 + l16] = o[nb][r] * inv;
  }
}

// ---------------------------------------------------------------------------
extern "C" void kernel_launch(void* const* d_in, const int* in_sizes, int n_in,
                              void* d_out, int out_size, void* d_ws, size_t ws_size,
                              hipStream_t stream) {
  (void)in_sizes; (void)n_in; (void)out_size; (void)ws_size;
  const float* x  = (const float*)d_in[0];   // [B,S,1024]
  const float* W0 = (const float*)d_in[1];   // [1024,3072]
  const float* b0 = (const float*)d_in[2];   // [3072]
  float* out = (float*)d_out;                // [B,S,1024] == [B][H][S][64] flat

  const size_t per = (size_t)Bb * H * S * HD;  // 4M floats = 16 MB each
  float* Qb  = (float*)d_ws;
  float* Ktp = Qb + per;
  float* Vp  = Ktp + per;                      // total 48 MB workspace

  dim3 g1(N3 / 64, (Bb * S) / 128);            // 48 x 32 blocks
  qkv_gemm<<<g1, 256, 0, stream>>>(x, W0, b0, Qb, Ktp, Vp);

  dim3 g2(Bb * H * (S / 128));                 // 512 blocks, 8 waves each
  attn<<<g2, 256, 0, stream>>>(Qb, Ktp, Vp, out);
}